// _ProposalLayer_62861141344464
// MI455X (gfx1250) — compile-verified
//
#include <hip/hip_runtime.h>
#include <hip/hip_bf16.h>
#include <stdint.h>

#define Bn 4
#define An 9
#define Hn 128
#define Wn 128
#define NPER (An * Hn * Wn)        // 147456 boxes per batch
#define PRE_TOPN 4000
#define POST_TOPN 1000
#define NMS_TH 0.7f
#define MASK_WORDS 125             // 4000 / 32
#define ROW_TILES 250              // 4000 / 16
#define SORTN 8192
#define HBINS 4096

typedef float v2f  __attribute__((ext_vector_type(2)));
typedef float v8f  __attribute__((ext_vector_type(8)));
typedef unsigned int v4u __attribute__((ext_vector_type(4)));
typedef int v8i  __attribute__((ext_vector_type(8)));
typedef int v4i  __attribute__((ext_vector_type(4)));

// ---------------------------------------------------------------- anchors ---
__device__ __forceinline__ void anchor_of(int a, float& x1, float& y1, float& x2, float& y2) {
    const float ratios[3] = {0.5f, 1.0f, 2.0f};
    const float scales[3] = {8.0f, 16.0f, 32.0f};
    float r = ratios[a / 3];
    float s = scales[a % 3];
    float ws = roundf(sqrtf(256.0f / r));
    float hs = roundf(ws * r);
    float Wf = ws * s, Hf = hs * s;
    x1 = 7.5f - 0.5f * (Wf - 1.0f);
    y1 = 7.5f - 0.5f * (Hf - 1.0f);
    x2 = 7.5f + 0.5f * (Wf - 1.0f);
    y2 = 7.5f + 0.5f * (Hf - 1.0f);
}

// decode + clip one proposal; (b, a, h, w) indexing into bbox_deltas (B,36,H,W)
__device__ __forceinline__ float4 decode_box(const float* __restrict__ deltas,
                                             int b, int a, int h, int w,
                                             float hmax, float wmax) {
    float ax1, ay1, ax2, ay2;
    anchor_of(a, ax1, ay1, ax2, ay2);
    float sx = (float)w * 16.0f, sy = (float)h * 16.0f;
    ax1 += sx; ax2 += sx; ay1 += sy; ay2 += sy;
    float aw = ax2 - ax1 + 1.0f;
    float ah = ay2 - ay1 + 1.0f;
    float cx = ax1 + 0.5f * aw;
    float cy = ay1 + 0.5f * ah;
    size_t base = (((size_t)b * 36 + 4 * a) * (Hn * Wn)) + (size_t)h * Wn + w;
    float dx = deltas[base];
    float dy = deltas[base + (size_t)(Hn * Wn)];
    float dw = deltas[base + (size_t)2 * (Hn * Wn)];
    float dh = deltas[base + (size_t)3 * (Hn * Wn)];
    float pcx = dx * aw + cx;
    float pcy = dy * ah + cy;
    float pw  = expf(dw) * aw;
    float ph  = expf(dh) * ah;
    float bx1 = pcx - 0.5f * pw;
    float by1 = pcy - 0.5f * ph;
    float bx2 = pcx + 0.5f * pw;
    float by2 = pcy + 0.5f * ph;
    bx1 = fminf(fmaxf(bx1, 0.0f), wmax);
    by1 = fminf(fmaxf(by1, 0.0f), hmax);
    bx2 = fminf(fmaxf(bx2, 0.0f), wmax);
    by2 = fminf(fmaxf(by2, 0.0f), hmax);
    return make_float4(bx1, by1, bx2, by2);
}

// ------------------------------------------------- kernel 1: masked scores ---
__global__ void k_scores(const float* __restrict__ scores,
                         const float* __restrict__ deltas,
                         const float* __restrict__ im_info,
                         const float* __restrict__ valid_range,
                         float* __restrict__ scoresW) {
    int t = blockIdx.x * blockDim.x + threadIdx.x;
    if (t >= Bn * NPER) return;
    int b = t / NPER;
    int n = t - b * NPER;
    int a = n % An;
    int pix = n / An;
    int w = pix % Wn;
    int h = pix / Wn;
    float hmax = im_info[b * 3 + 0] - 1.0f;
    float wmax = im_info[b * 3 + 1] - 1.0f;
    float4 bx = decode_box(deltas, b, a, h, w, hmax, wmax);
    float wd = bx.z - bx.x;
    float ht = bx.w - bx.y;
    float area = wd * ht;
    float mn = valid_range[b * 2 + 0]; mn *= mn;
    float mx = valid_range[b * 2 + 1]; mx *= mx;
    float sc = scores[(((size_t)b * (2 * An) + An + a) * (Hn * Wn)) + (size_t)h * Wn + w];
    if (area < mn || area > mx) sc = -1.0f;
    scoresW[t] = sc;
}

// ----------------------------------------------- kernel 2: top-4000 select ---
__device__ __forceinline__ unsigned int key_of(float s) {
    unsigned int u = __float_as_uint(s);
    return (u & 0x80000000u) ? ~u : (u | 0x80000000u);
}

extern __shared__ char smem_dyn[];

__global__ __launch_bounds__(1024) void k_topk(const float* __restrict__ scoresW,
                                               unsigned int* __restrict__ topIdx) {
    unsigned long long* cand = (unsigned long long*)smem_dyn;          // 64 KB
    unsigned int* hist = (unsigned int*)(smem_dyn + SORTN * 8);        // 16 KB
    __shared__ unsigned int s_T, s_counter;
    const int tid = threadIdx.x;
    const int b = blockIdx.x;
    const float* sc = scoresW + (size_t)b * NPER;

    for (int i = tid; i < HBINS; i += 1024) hist[i] = 0u;
    __syncthreads();
    for (int i = tid; i < NPER; i += 1024)
        atomicAdd(&hist[key_of(sc[i]) >> 20], 1u);
    __syncthreads();
    if (tid == 0) {
        unsigned int cum = 0; int T = 0;
        for (int bin = HBINS - 1; bin >= 0; --bin) {
            cum += hist[bin];
            if (cum >= PRE_TOPN) { T = bin; break; }
        }
        s_T = (unsigned int)T;
        s_counter = 0u;
    }
    __syncthreads();
    unsigned int T = s_T;
    for (int i = tid; i < SORTN; i += 1024) cand[i] = 0ull;
    __syncthreads();
    for (int i = tid; i < NPER; i += 1024) {
        unsigned int k = key_of(sc[i]);
        if ((k >> 20) >= T) {
            unsigned int pos = atomicAdd(&s_counter, 1u);
            if (pos < SORTN)
                cand[pos] = ((unsigned long long)k << 32) | (unsigned long long)(~(unsigned int)i);
        }
    }
    __syncthreads();
    // bitonic sort, descending (ties: larger ~idx first == smaller idx first)
    for (int k = 2; k <= SORTN; k <<= 1) {
        for (int j = k >> 1; j > 0; j >>= 1) {
            for (int i = tid; i < SORTN; i += 1024) {
                int ixj = i ^ j;
                if (ixj > i) {
                    unsigned long long va = cand[i], vb = cand[ixj];
                    bool desc = ((i & k) == 0);
                    if (desc ? (va < vb) : (va > vb)) { cand[i] = vb; cand[ixj] = va; }
                }
            }
            __syncthreads();
        }
    }
    for (int r = tid; r < PRE_TOPN; r += 1024) {
        unsigned long long p = cand[r];
        unsigned int idx = (p == 0ull) ? 0u : ~((unsigned int)(p & 0xFFFFFFFFull));
        if (idx >= NPER) idx = 0u;
        topIdx[(size_t)b * PRE_TOPN + r] = idx;
    }
}

// ------------------------------------------ kernel 3: gather selected boxes ---
__global__ void k_gather(const unsigned int* __restrict__ topIdx,
                         const float* __restrict__ deltas,
                         const float* __restrict__ im_info,
                         float4* __restrict__ boxesW) {
    int t = blockIdx.x * blockDim.x + threadIdx.x;
    if (t >= Bn * PRE_TOPN) return;
    int b = t / PRE_TOPN;
    unsigned int idx = topIdx[t];
    int a = idx % An;
    int pix = idx / An;
    int w = pix % Wn;
    int h = pix / Wn;
    float hmax = im_info[b * 3 + 0] - 1.0f;
    float wmax = im_info[b * 3 + 1] - 1.0f;
    boxesW[t] = decode_box(deltas, b, a, h, w, hmax, wmax);
}

// ---------------------- kernel 4: suppression bitmask (TDM + WMMA + VALU) ---
__global__ __launch_bounds__(32) void k_mask(const float4* __restrict__ boxesW,
                                             unsigned int* __restrict__ maskW) {
    __shared__ float4 colBoxS[32];
    __shared__ float4 rowBoxS[16];
    __shared__ float  rowAreaS[16];
    __shared__ float  colAreaS[32];
    __shared__ unsigned int rowWord[16];

    const int lane = threadIdx.x;
    const int b = blockIdx.z;
    const int colStart = blockIdx.x * 32;
    const int rowStart = blockIdx.y * 16;

    // --- TDM: DMA the 32 column boxes (512 contiguous bytes) into LDS -------
    {
        const char* gptr = (const char*)(boxesW + (size_t)b * PRE_TOPN + colStart);
        uint64_t ga = (uint64_t)(uintptr_t)gptr;
        uint32_t ldsOff = (uint32_t)(uintptr_t)(&colBoxS[0]);  // flat LDS addr low 32 = LDS offset
        v4u g0 = { 0x1u,                       // count = 1 valid descriptor
                   ldsOff,                     // lds_addr
                   (uint32_t)ga,               // global_addr[31:0]
                   (uint32_t)((ga >> 32) & 0x01FFFFFFu) | 0x80000000u };  // ga[56:32] | type=2
        v8i g1 = { (int)(3u << 16),            // data_size = 3 (8B units)
                   (int)(64u << 16),           // tensor_dim0 = 64 (low16 @ bits 63:48)
                   (int)(1u << 16),            // tensor_dim1 = 1  (low16 @ bits 111:80)
                   (int)(64u << 16),           // tile_dim0 = 64   (bits 127:112)
                   (int)1,                     // tile_dim1 = 1    (bits 143:128)
                   (int)64,                    // tensor_dim0_stride = 64 (bits 207:160)
                   0, 0 };
        v4i g2 = {0, 0, 0, 0};
        v4i g3 = {0, 0, 0, 0};
        v8i g4 = {0, 0, 0, 0, 0, 0, 0, 0};     // unused group (6-arg builtin form)
        __builtin_amdgcn_tensor_load_to_lds(g0, g1, g2, g3, g4, 0);
    }

    // --- row boxes via vector loads, per-tile areas -------------------------
    if (lane < 16) {
        float4 rb = boxesW[(size_t)b * PRE_TOPN + rowStart + lane];
        rowBoxS[lane] = rb;
        rowAreaS[lane] = (rb.z - rb.x + 1.0f) * (rb.w - rb.y + 1.0f);
        rowWord[lane] = 0u;
    }
    __builtin_amdgcn_s_wait_tensorcnt(0);
    __syncthreads();
    {
        float4 cb = colBoxS[lane];
        colAreaS[lane] = (cb.z - cb.x + 1.0f) * (cb.w - cb.y + 1.0f);
    }
    __syncthreads();

    // --- WMMA: outer-sum areas[i] + areas[j] for both 16x16 half-tiles ------
    // A (16x4) = [area_row, 1, 0, 0]; B (4x16) = [ones; area_col; 0; 0]
    v2f a, b0, b1;
    if (lane < 16) {
        a.x = rowAreaS[lane];        a.y = 1.0f;
        b0.x = 1.0f;                 b0.y = colAreaS[lane];
        b1.x = 1.0f;                 b1.y = colAreaS[lane + 16];
    } else {
        a.x = 0.0f;  a.y = 0.0f;
        b0.x = 0.0f; b0.y = 0.0f;
        b1.x = 0.0f; b1.y = 0.0f;
    }
    v8f cz = {};
    v8f sum0 = __builtin_amdgcn_wmma_f32_16x16x4_f32(false, a, false, b0, (short)0, cz, false, false);
    v8f sum1 = __builtin_amdgcn_wmma_f32_16x16x4_f32(false, a, false, b1, (short)0, cz, false, false);

    // --- VALU intersection + threshold, assemble 32-bit row words -----------
    const int half = lane >> 4;
    const int n0 = lane & 15;
    float4 cbA = colBoxS[n0];
    float4 cbB = colBoxS[n0 + 16];
    #pragma unroll
    for (int v = 0; v < 8; ++v) {
        int m = v + 8 * half;
        int gRow = rowStart + m;
        float4 rb = rowBoxS[m];
        // half-tile 0 (cols colStart + n0)
        {
            float iw = fminf(rb.z, cbA.z) - fmaxf(rb.x, cbA.x) + 1.0f;
            float ih = fminf(rb.w, cbA.w) - fmaxf(rb.y, cbA.y) + 1.0f;
            float inter = fmaxf(iw, 0.0f) * fmaxf(ih, 0.0f);
            bool sup = (inter > NMS_TH * (sum0[v] - inter)) && ((colStart + n0) > gRow);
            if (sup) atomicOr(&rowWord[m], 1u << n0);
        }
        // half-tile 1 (cols colStart + 16 + n0)
        {
            float iw = fminf(rb.z, cbB.z) - fmaxf(rb.x, cbB.x) + 1.0f;
            float ih = fminf(rb.w, cbB.w) - fmaxf(rb.y, cbB.y) + 1.0f;
            float inter = fmaxf(iw, 0.0f) * fmaxf(ih, 0.0f);
            bool sup = (inter > NMS_TH * (sum1[v] - inter)) && ((colStart + 16 + n0) > gRow);
            if (sup) atomicOr(&rowWord[m], 1u << (n0 + 16));
        }
    }
    __syncthreads();
    if (lane < 16)
        maskW[((size_t)b * PRE_TOPN + rowStart + lane) * MASK_WORDS + blockIdx.x] = rowWord[lane];
}

// ------------------------------- kernel 5: greedy scan + compact + output ---
__global__ __launch_bounds__(128) void k_nms(const unsigned int* __restrict__ maskW,
                                             const float4* __restrict__ boxesW,
                                             float* __restrict__ out) {
    __shared__ unsigned int remv[MASK_WORDS];
    __shared__ unsigned int keepList[POST_TOPN];
    __shared__ unsigned int keepCount;
    const int tid = threadIdx.x;
    const int b = blockIdx.x;
    if (tid < MASK_WORDS) remv[tid] = 0u;
    if (tid == 0) keepCount = 0u;
    __syncthreads();

    for (int i = 0; i < PRE_TOPN; ++i) {
        bool keep = ((remv[i >> 5] >> (i & 31)) & 1u) == 0u;
        __syncthreads();
        if (keep) {
            if (tid < MASK_WORDS)
                remv[tid] |= maskW[((size_t)b * PRE_TOPN + i) * MASK_WORDS + tid];
            if (tid == 0) {
                unsigned int kc = keepCount;
                if (kc < POST_TOPN) keepList[kc] = (unsigned int)i;
                keepCount = kc + 1u;
            }
        }
        __syncthreads();
    }

    unsigned int kc = keepCount;
    for (int r = tid; r < POST_TOPN; r += 128) {
        float* o = out + ((size_t)b * POST_TOPN + r) * 5;
        o[0] = (float)b;
        if ((unsigned int)r < kc) {
            float4 bx = boxesW[(size_t)b * PRE_TOPN + keepList[r]];
            o[1] = bx.x; o[2] = bx.y; o[3] = bx.z; o[4] = bx.w;
        } else {
            o[1] = 0.0f; o[2] = 0.0f; o[3] = 0.0f; o[4] = 0.0f;
        }
    }
}

// ---------------------------------------------------------------------------
extern "C" void kernel_launch(void* const* d_in, const int* in_sizes, int n_in,
                              void* d_out, int out_size, void* d_ws, size_t ws_size,
                              hipStream_t stream) {
    (void)in_sizes; (void)n_in; (void)out_size; (void)ws_size;
    const float* scores      = (const float*)d_in[0];
    const float* bbox_deltas = (const float*)d_in[1];
    const float* im_info     = (const float*)d_in[2];
    const float* valid_range = (const float*)d_in[3];
    float* out = (float*)d_out;

    char* ws = (char*)d_ws;
    size_t off = 0;
    auto alloc = [&](size_t bytes) { char* p = ws + off; off = (off + bytes + 255) & ~(size_t)255; return p; };
    float*        scoresW = (float*)alloc((size_t)Bn * NPER * sizeof(float));
    unsigned int* topIdx  = (unsigned int*)alloc((size_t)Bn * PRE_TOPN * sizeof(unsigned int));
    float4*       boxesW  = (float4*)alloc((size_t)Bn * PRE_TOPN * sizeof(float4));
    unsigned int* maskW   = (unsigned int*)alloc((size_t)Bn * PRE_TOPN * MASK_WORDS * sizeof(unsigned int));

    // 1) decode + masked scores
    {
        int total = Bn * NPER;
        k_scores<<<(total + 255) / 256, 256, 0, stream>>>(scores, bbox_deltas, im_info, valid_range, scoresW);
    }
    // 2) exact per-batch top-4000 (histogram threshold + bitonic sort in LDS)
    {
        size_t shmem = (size_t)SORTN * 8 + (size_t)HBINS * 4;  // 80 KB dynamic LDS (WGP has 320 KB)
        k_topk<<<Bn, 1024, shmem, stream>>>(scoresW, topIdx);
    }
    // 3) gather selected boxes
    {
        int total = Bn * PRE_TOPN;
        k_gather<<<(total + 255) / 256, 256, 0, stream>>>(topIdx, bbox_deltas, im_info, boxesW);
    }
    // 4) suppression bitmask: TDM column staging + WMMA area outer-sum
    {
        dim3 grid(MASK_WORDS, ROW_TILES, Bn);
        k_mask<<<grid, 32, 0, stream>>>(boxesW, maskW);
    }
    // 5) sequential greedy NMS + output
    {
        k_nms<<<Bn, 128, 0, stream>>>(maskW, boxesW, out);
    }
}